// NMPC_8057358647378
// MI455X (gfx1250) — compile-verified
//
#include <hip/hip_runtime.h>

typedef __attribute__((ext_vector_type(2))) float v2f;
typedef __attribute__((ext_vector_type(8))) float v8f;

#define NXD    128
#define NSTEPS 4096

// Single-WGP latency-optimized sequential rollout.
// 256 threads = 8 waves; wave w owns rows [16w, 16w+16) of A, held in VGPRs
// in the native V_WMMA_F32_16X16X4_F32 A-matrix layout:
//   lane L (L%16 = m-offset, L/16 selects K-pair), a_reg[kb] = A[16w+L%16][4kb + 2*(L/16) + {0,1}]
__launch_bounds__(256, 1)
__global__ void nmpc_rollout_kernel(const float* __restrict__ x0,
                                    const float* __restrict__ x_ref,
                                    const float* __restrict__ u_seq,
                                    const float* __restrict__ A,
                                    const float* __restrict__ Bm,
                                    float* __restrict__ out)
{
    __shared__ float  s_x[NXD];     // current state
    __shared__ float  s_y[NXD];     // A @ x
    __shared__ float  s_B0[NXD];    // B column 0
    __shared__ float  s_B1[NXD];    // B column 1
    __shared__ double s_red[256];   // final reduction

    const int tid  = threadIdx.x;
    const int lane = tid & 31;
    const int wave = tid >> 5;      // 0..7 -> row block
    const int half = lane >> 4;     // 0: K-pair {0,1}; 1: K-pair {2,3}
    const int l16  = lane & 15;     // row within block (A) / column (C,D)

    // ---- Load this wave's 16x128 slice of A into registers (once) ----
    v2f a_reg[32];
    {
        const int row = wave * 16 + l16;
        const float* ap = A + (size_t)row * NXD + half * 2;
#pragma unroll
        for (int kb = 0; kb < 32; ++kb) {
            a_reg[kb].x = ap[kb * 4 + 0];
            a_reg[kb].y = ap[kb * 4 + 1];
        }
    }

    // ---- Stage x0, B into LDS; t=0 error term ----
    double err = 0.0;
    if (tid < NXD) {
        s_x[tid]  = x0[tid];
        s_B0[tid] = Bm[tid * 2 + 0];
        s_B1[tid] = Bm[tid * 2 + 1];
        float d = x0[tid] - x_ref[tid];
        err = (double)(d * d);
    }
    __syncthreads();

    for (int t = 0; t < NSTEPS; ++t) {
        // Prefetch x_ref a few steps ahead (gfx1250 global_prefetch_b8).
        if (tid < NXD) {
            int tp = (t + 8 <= NSTEPS) ? (t + 8) : NSTEPS;
            __builtin_prefetch(&x_ref[(size_t)tp * NXD + tid], 0, 1);
        }

        // ---- y = A @ x : 32 chained fp32 WMMAs, K = 0..127 in steps of 4.
        // B operand: x replicated across all 16 columns, so every column of D
        // equals y. Per the 4x16 B layout, lanes 0-15 supply rows K={4k,4k+1},
        // lanes 16-31 supply K={4k+2,4k+3}.
        v8f acc = {0.f, 0.f, 0.f, 0.f, 0.f, 0.f, 0.f, 0.f};
        const float* xb = &s_x[half * 2];
#pragma unroll
        for (int kb = 0; kb < 32; ++kb) {
            v2f bx;
            bx.x = xb[kb * 4 + 0];
            bx.y = xb[kb * 4 + 1];
            acc = __builtin_amdgcn_wmma_f32_16x16x4_f32(
                /*neg_a=*/false, a_reg[kb],
                /*neg_b=*/false, bx,
                /*c_mod=*/(short)0, acc,
                /*reuse_a=*/false, /*reuse_b=*/false);
        }

        // Column N=0 lanes (lane%16==0) hold y: VGPR r <-> row 16w + 8*half + r.
        if (l16 == 0) {
#pragma unroll
            for (int r = 0; r < 8; ++r)
                s_y[wave * 16 + half * 8 + r] = acc[r];
        }
        __syncthreads();

        // ---- State update + error accumulation (threads 0..127) ----
        if (tid < NXD) {
            float u0  = u_seq[t];             // u_sequence (2, N) row-major
            float u1  = u_seq[NSTEPS + t];
            float pre = s_y[tid] + s_B0[tid] * u0 + s_B1[tid] * u1;
            float xn  = s_x[tid] + tanhf(pre);
            s_x[tid]  = xn;
            float d   = xn - x_ref[(size_t)(t + 1) * NXD + tid];
            err += (double)(d * d);
        }
        __syncthreads();
    }

    // ---- MSE reduction over the 128 state columns ----
    s_red[tid] = err;
    __syncthreads();
    for (int s = 128; s > 0; s >>= 1) {
        if (tid < s) s_red[tid] += s_red[tid + s];
        __syncthreads();
    }
    if (tid == 0)
        out[0] = (float)(s_red[0] / (double)((NSTEPS + 1) * NXD));
}

extern "C" void kernel_launch(void* const* d_in, const int* in_sizes, int n_in,
                              void* d_out, int out_size, void* d_ws, size_t ws_size,
                              hipStream_t stream) {
    (void)in_sizes; (void)n_in; (void)out_size; (void)d_ws; (void)ws_size;
    const float* x0    = (const float*)d_in[0];
    const float* x_ref = (const float*)d_in[1];
    const float* u_seq = (const float*)d_in[2];
    const float* A     = (const float*)d_in[3];
    const float* B     = (const float*)d_in[4];
    float* out = (float*)d_out;

    // Strictly sequential scan: exactly one workgroup (8 wave32 on one WGP).
    nmpc_rollout_kernel<<<1, 256, 0, stream>>>(x0, x_ref, u_seq, A, B, out);
}